// Yolov3Layer_86552180949072
// MI455X (gfx1250) — compile-verified
//
#include <hip/hip_runtime.h>
#include <stdint.h>

// YOLOv3 detection head for MI455X (gfx1250).
// Pure bandwidth-bound transpose + activation: 377 MB traffic -> ~16us at 23.3 TB/s.
// Strategy: TDM (tensor_load_to_lds) pulls an 85x128 strided tile (row stride 5776
// floats) into LDS with hardware padding (stride 129 dwords -> bank-conflict-free),
// then 256 threads emit coalesced float4 stores with fused sigmoid/exp transforms.

#define NA     3
#define ATTRS  85
#define GRID_  76
#define HW     (GRID_ * GRID_)        // 5776
#define TILE_S 128
#define LDSTR  (TILE_S + 1)           // 129, via TDM pad (1 dword every 128 dwords)
#define NTILES ((HW + TILE_S - 1) / TILE_S)   // 46 (last tile: 16 valid)

typedef unsigned int u32x4 __attribute__((ext_vector_type(4)));
typedef int          i32x8 __attribute__((ext_vector_type(8)));
typedef int          i32x4 __attribute__((ext_vector_type(4)));

#define HAVE_TDM (__has_builtin(__builtin_amdgcn_tensor_load_to_lds) && \
                  __has_builtin(__builtin_amdgcn_s_wait_tensorcnt))

__device__ __forceinline__ float fast_sigmoid(float x) {
    return __builtin_amdgcn_rcpf(1.0f + __expf(-x));
}

__global__ __launch_bounds__(256) void yolov3_head_kernel(
        const float* __restrict__ in,      // (B, 255, 76, 76)
        const float* __restrict__ anchors, // (3, 2)
        float* __restrict__ out)           // (B, 3*5776, 85)
{
    __shared__ float tile[ATTRS * LDSTR];  // 43,860 B

    const int tid = threadIdx.x;
    const int t   = blockIdx.x;            // spatial tile   [0, 46)
    const int a   = blockIdx.y;            // anchor         [0, 3)
    const int b   = blockIdx.z;            // batch          [0, 32)
    const int s0  = t * TILE_S;
    const int rem = HW - s0;
    const int nvalid = rem < TILE_S ? rem : TILE_S;

    const size_t inbase = ((size_t)(b * (NA * ATTRS) + a * ATTRS)) * HW + s0;

    // Keep the compiler honest about `tile` being written (TDM writes it via
    // a descriptor, invisible to alias analysis). Never executed.
    if (t == 0x7fffffff) tile[tid] = anchors[0];

#if HAVE_TDM
    if (tid < 32) {  // wave 0 issues one TDM transfer for the whole 85x128 tile
        const uint64_t gaddr   = (uint64_t)(uintptr_t)(in + inbase);
        const uint32_t ldsaddr = (uint32_t)(uintptr_t)(&tile[0]);

        u32x4 g0;
        g0[0] = 1u;                                         // count=1 (user mode)
        g0[1] = ldsaddr;                                    // lds_addr (bytes)
        g0[2] = (uint32_t)(gaddr & 0xFFFFFFFFu);            // global_addr[31:0]
        g0[3] = (uint32_t)((gaddr >> 32) & 0x01FFFFFFu)     // global_addr[56:32]
              | (2u << 30);                                 // type = 2 ("image")

        i32x8 g1;
        g1[0] = (2 << 16)        // data_size = 4 bytes
              | (1 << 20)        // pad_enable: pad LDS dest
              | (6 << 22);       // pad_interval: every 128 DWORDs (pad_amount=0 -> 1 DWORD)
        g1[1] = (rem & 0xFFFF) << 16;        // tensor_dim0 = remaining (OOB x -> zero-fill)
        g1[2] = (ATTRS & 0xFFFF) << 16;      // tensor_dim1 = 85
        g1[3] = (TILE_S & 0xFFFF) << 16;     // tile_dim0 = 128
        g1[4] = ATTRS;                       // tile_dim1 = 85 (tile_dim2 = 0)
        g1[5] = HW;                          // tensor_dim0_stride = 5776 (lo 32)
        g1[6] = 0;                           // stride hi, tensor_dim1_stride lo
        g1[7] = 0;

        i32x4 gz4 = {0, 0, 0, 0};            // groups 2/3 unused for 2D tiles
        i32x8 gz8 = {0, 0, 0, 0, 0, 0, 0, 0};
        __builtin_amdgcn_tensor_load_to_lds(g0, g1, gz4, gz4, gz8, 0);
        __builtin_amdgcn_s_wait_tensorcnt(0);
    }
    __syncthreads();
#else
    // Fallback: cooperative coalesced staging (consecutive lanes -> consecutive s).
    for (int idx = tid; idx < ATTRS * TILE_S; idx += 256) {
        const int s    = idx & (TILE_S - 1);
        const int attr = idx >> 7;
        float v = 0.0f;
        if (s < nvalid) v = in[inbase + (size_t)attr * HW + s];
        tile[attr * LDSTR + s] = v;
    }
    __syncthreads();
#endif

    // Per-anchor scales (uniform scalar loads).
    const float aw = anchors[a * 2 + 0] * (1.0f / 608.0f);
    const float ah = anchors[a * 2 + 1] * (1.0f / 608.0f);

    float* outp = out + ((size_t)(b * NA + a) * HW + s0) * ATTRS;
    const int nelems = nvalid * ATTRS;       // multiple of 4 (nvalid in {128,16})

    for (int base = tid * 4; base < nelems; base += 256 * 4) {
        float4 v;
        float* vp = (float*)&v;
#pragma unroll
        for (int k = 0; k < 4; ++k) {
            const int e    = base + k;
            const int s    = (unsigned)e / ATTRS;
            const int attr = e - s * ATTRS;
            const float x  = tile[attr * LDSTR + s];
            const int s_abs = s0 + s;
            float r;
            if (attr == 0) {
                r = ((float)(s_abs % GRID_) + fast_sigmoid(x)) * (1.0f / GRID_);
            } else if (attr == 1) {
                r = ((float)(s_abs / GRID_) + fast_sigmoid(x)) * (1.0f / GRID_);
            } else if (attr == 2) {
                r = __expf(x) * aw;
            } else if (attr == 3) {
                r = __expf(x) * ah;
            } else {
                r = fast_sigmoid(x);
            }
            vp[k] = r;
        }
        *(float4*)(outp + base) = v;   // 16B-aligned, fully coalesced
    }
}

extern "C" void kernel_launch(void* const* d_in, const int* in_sizes, int n_in,
                              void* d_out, int out_size, void* d_ws, size_t ws_size,
                              hipStream_t stream) {
    const float* in      = (const float*)d_in[0];
    const float* anchors = (const float*)d_in[1];
    float* out           = (float*)d_out;

    (void)in_sizes; (void)n_in; (void)out_size; (void)d_ws; (void)ws_size;
    dim3 grid(NTILES, NA, /*B=*/in_sizes[0] / (NA * ATTRS * HW));
    yolov3_head_kernel<<<grid, 256, 0, stream>>>(in, anchors, out);
}